// MyModel_77644418777127
// MI455X (gfx1250) — compile-verified
//
#include <hip/hip_runtime.h>
#include <hip/hip_bf16.h>

#define N_U 20000
#define N_I 30000
#define NN  50000
#define DD  64
#define HH  4
#define DKK 16
#define RR  3
#define EE  1000000
#define TT  768

// output layout (floats)
#define OFF_UG  0ull
#define OFF_IG  1280000ull
#define OFF_REL 3200064ull
#define OFF_UA  3200256ull
#define OFF_IA  4480256ull
#define OFF_HU  6400320ull
#define OFF_HI  7680320ull

typedef float v2f __attribute__((ext_vector_type(2)));
typedef float v8f __attribute__((ext_vector_type(8)));

// ---------------------------------------------------------------------------
// Generic fp32 WMMA GEMM:  C[M x 64] = ep( A[M x K] @ (diag(kscale?) W[K x 64]) + bias )
// ep: 0 = none, 1 = relu, 2 = leaky_relu(0.01)
// Wave32: each wave computes a 16-row strip x 64 cols (4 x v8f accumulators),
// K-loop in 64-wide chunks staged into LDS (transposed, pitch 66 to spread banks).
// M must be a multiple of 16 (20000/30000/50000 all are), K a multiple of 64.
// ---------------------------------------------------------------------------
#define LDSP 66
__global__ __launch_bounds__(256)
void gemm_f32_wmma(const float* __restrict__ A, const float* __restrict__ W,
                   const float* __restrict__ bias, const float* __restrict__ kscale,
                   float* __restrict__ C, int M, int K, int ep)
{
    __shared__ float wls[64 * LDSP];
    const int lane = threadIdx.x & 31;
    const int wave = threadIdx.x >> 5;
    const int m0   = (blockIdx.x * 8 + wave) * 16;     // strip base row (wave-uniform)
    const bool active = (m0 < M);
    const int mrow = m0 + (lane & 15);                 // A row for this lane
    const int kb   = (lane >> 4) << 1;                 // K sub-offset {0,2} per ISA A layout
    const int ncol = lane & 15;                        // column within 16-wide tile

    v8f acc[4] = {};
    const int nkc = K >> 6;
    for (int kc = 0; kc < nkc; ++kc) {
        __syncthreads();
        // stage W chunk [kc*64 .. +63] x 64 into LDS, transposed: wls[n][k]
        for (int idx = threadIdx.x; idx < 4096; idx += 256) {
            int k = idx >> 6, n = idx & 63;
            float wv = W[(size_t)(kc * 64 + k) * 64 + n];
            if (kscale) wv *= kscale[kc * 64 + k];
            wls[n * LDSP + k] = wv;
        }
        __syncthreads();
        if (active) {
            const float* arow = A + (size_t)mrow * K + (size_t)kc * 64;
            #pragma unroll
            for (int kk = 0; kk < 64; kk += 4) {
                // A fragment: 16x4 fp32 tile, lane (m = lane&15, k = kk + kb + {0,1})
                v2f a = *(const v2f*)(arow + kk + kb);
                #pragma unroll
                for (int nt = 0; nt < 4; ++nt) {
                    // B fragment: 4x16 tile, lane (n = ncol, k = kk + kb + {0,1})
                    v2f b = *(const v2f*)(&wls[(nt * 16 + ncol) * LDSP + kk + kb]);
                    acc[nt] = __builtin_amdgcn_wmma_f32_16x16x4_f32(
                        false, a, false, b, (short)0, acc[nt], false, false);
                }
            }
        }
    }
    if (active) {
        const int rbase = m0 + ((lane >> 4) << 3);     // C row base per ISA C/D layout
        #pragma unroll
        for (int nt = 0; nt < 4; ++nt) {
            int col = nt * 16 + ncol;
            float bs = bias ? bias[col] : 0.f;
            #pragma unroll
            for (int v = 0; v < 8; ++v) {
                float x = acc[nt][v] + bs;
                if (ep == 1)      x = fmaxf(x, 0.f);
                else if (ep == 2) x = (x > 0.f) ? x : 0.01f * x;
                C[(size_t)(rbase + v) * 64 + col] = x;
            }
        }
    }
}

// ---------------------------------------------------------------------------
// Edge-softmax phase. Monotonic float<->uint encoding for atomicMax.
// ---------------------------------------------------------------------------
__device__ __forceinline__ unsigned int fenc(float f) {
    unsigned int u = __float_as_uint(f);
    return (u & 0x80000000u) ? ~u : (u | 0x80000000u);
}
__device__ __forceinline__ float fdec(unsigned int u) {
    u = (u & 0x80000000u) ? (u & 0x7fffffffu) : ~u;
    return __uint_as_float(u);
}

__global__ __launch_bounds__(256)
void edge_scores_kernel(const int* __restrict__ rows, const int* __restrict__ cols,
                        const float* __restrict__ Q, const float* __restrict__ Km,
                        const float* __restrict__ rel, float* __restrict__ scores,
                        unsigned int* __restrict__ mx)
{
    int i = blockIdx.x * 256 + threadIdx.x;            // i = e*H + h
    if (i >= EE * HH) return;
    int e = i >> 2, h = i & 3;
    int r = rows[e], c = cols[e];
    const float4* q  = (const float4*)(Q  + (size_t)c * DD + h * DKK);
    const float4* k  = (const float4*)(Km + (size_t)r * DD + h * DKK);
    const float4* rv = (const float4*)(rel + h * DKK);
    float s = 0.f;
    #pragma unroll
    for (int j = 0; j < 4; ++j) {
        float4 a = q[j], b = k[j], t = rv[j];
        s += a.x * (b.x + t.x) + a.y * (b.y + t.y) + a.z * (b.z + t.z) + a.w * (b.w + t.w);
    }
    s *= 0.25f;                                        // 1/sqrt(DK)
    scores[i] = s;
    atomicMax(mx + (size_t)c * HH + h, fenc(s));
}

__global__ __launch_bounds__(256)
void edge_exp_kernel(const int* __restrict__ cols, float* __restrict__ scores,
                     const unsigned int* __restrict__ mx, float* __restrict__ den)
{
    int i = blockIdx.x * 256 + threadIdx.x;
    if (i >= EE * HH) return;
    int e = i >> 2, h = i & 3;
    int c = cols[e];
    float m = fdec(mx[(size_t)c * HH + h]);
    float v = __expf(scores[i] - m);
    scores[i] = v;
    atomicAdd(den + (size_t)c * HH + h, v);
}

__global__ __launch_bounds__(256)
void edge_scatter_kernel(const int* __restrict__ rows, const int* __restrict__ cols,
                         const float* __restrict__ scores, const float* __restrict__ den,
                         const float* __restrict__ V, float* __restrict__ out)
{
    int i = blockIdx.x * 256 + threadIdx.x;
    if (i >= EE * HH) return;
    int e = i >> 2, h = i & 3;
    int r = rows[e], c = cols[e];
    float attn = scores[i] / den[(size_t)c * HH + h];
    const float4* vv = (const float4*)(V + (size_t)r * DD + h * DKK);
    float* o = out + (size_t)c * DD + h * DKK;
    #pragma unroll
    for (int j = 0; j < 4; ++j) {
        float4 t = vv[j];
        atomicAdd(o + j * 4 + 0, attn * t.x);
        atomicAdd(o + j * 4 + 1, attn * t.y);
        atomicAdd(o + j * 4 + 2, attn * t.z);
        atomicAdd(o + j * 4 + 3, attn * t.w);
    }
}

// ---------------------------------------------------------------------------
// Final combine: all_emb = 0.5*(ego + sum coeff[i]*part[i]); split into u/i outputs.
// ---------------------------------------------------------------------------
__global__ __launch_bounds__(256)
void combine_kernel(const float* __restrict__ ego, const float* __restrict__ p0,
                    const float* __restrict__ p1, const float* __restrict__ p2,
                    const float* __restrict__ p3, const float* __restrict__ coeff,
                    float* __restrict__ out)
{
    long i = (long)blockIdx.x * 256 + threadIdx.x;
    if (i >= (long)NN * DD) return;
    int n = (int)(i >> 6), d = (int)(i & 63);
    float c0 = coeff[0], c1 = coeff[1], c2 = coeff[2];
    float base = c0 * p0[i] + c1 * p1[i];
    float emb  = base + c2 * p2[i];
    float emba = base + c2 * p3[i];
    float eg = ego[i];
    float a = 0.5f * (eg + emb);
    float b = 0.5f * (eg + emba);
    if (n < N_U) {
        out[OFF_UG + (size_t)n * DD + d] = a;
        out[OFF_UA + (size_t)n * DD + d] = b;
    } else {
        size_t ni = (size_t)(n - N_U);
        out[OFF_IG + ni * DD + d] = a;
        out[OFF_IA + ni * DD + d] = b;
    }
}

// rel_out rows + zero token rows of i_g / i_a
__global__ void tail_kernel(const float* __restrict__ rel_emb, const float* __restrict__ W_rel,
                            float* __restrict__ out)
{
    int t = threadIdx.x;                               // 192 threads
    if (t < DD) {
        out[OFF_IG + (size_t)N_I * DD + t] = 0.f;
        out[OFF_IA + (size_t)N_I * DD + t] = 0.f;
    }
    if (t < RR * DD) {
        int r = t >> 6, d = t & 63;
        float s = 0.f;
        for (int k = 0; k < DD; ++k) s += rel_emb[r * DD + k] * W_rel[k * DD + d];
        out[OFF_REL + t] = 0.5f * (rel_emb[t] + s);
    }
}

// ---------------------------------------------------------------------------
extern "C" void kernel_launch(void* const* d_in, const int* in_sizes, int n_in,
                              void* d_out, int out_size, void* d_ws, size_t ws_size,
                              hipStream_t stream)
{
    const float* user_emb  = (const float*)d_in[0];
    const float* item_emb  = (const float*)d_in[1];
    const float* rel_emb   = (const float*)d_in[2];
    const float* W_gc      = (const float*)d_in[3];
    const float* W_rel     = (const float*)d_in[4];
    const float* coeff     = (const float*)d_in[5];
    const float* Wu1       = (const float*)d_in[6];
    const float* bu1       = (const float*)d_in[7];
    const float* Wu2       = (const float*)d_in[8];
    const float* bu2       = (const float*)d_in[9];
    const float* Wi1       = (const float*)d_in[10];
    const float* bi1       = (const float*)d_in[11];
    const float* Wi2       = (const float*)d_in[12];
    const float* bi2       = (const float*)d_in[13];
    const float* Wq        = (const float*)d_in[14];
    const float* Wk        = (const float*)d_in[15];
    const float* Wv        = (const float*)d_in[16];
    const float* Wo        = (const float*)d_in[17];
    const float* bq        = (const float*)d_in[18];
    const float* bk        = (const float*)d_in[19];
    const float* bv        = (const float*)d_in[20];
    const float* bo        = (const float*)d_in[21];
    const float* user_text = (const float*)d_in[22];
    const float* item_text = (const float*)d_in[23];
    const int*   edges     = (const int*)d_in[24];     // (R,2,E)
    const int*   aug       = (const int*)d_in[25];     // (2,E)
    float* out = (float*)d_out;

    // ---- workspace carve (~158 MB) ----
    char* wsp = (char*)d_ws;
    auto carve = [&](size_t bytes) {
        void* p = (void*)wsp;
        wsp += (bytes + 255) & ~(size_t)255;
        return p;
    };
    const size_t NDb = (size_t)NN * DD * sizeof(float);
    float* text     = (float*)carve(NDb);
    float* ego      = (float*)carve(NDb);
    float* qbuf     = (float*)carve(NDb);
    float* kbuf     = (float*)carve(NDb);
    float* vbuf     = (float*)carve(NDb);
    float* agg      = (float*)carve(NDb);
    float* attnproj = (float*)carve(NDb);
    float* part[4];
    for (int i = 0; i < 4; ++i) part[i] = (float*)carve(NDb);
    float* scores   = (float*)carve((size_t)EE * HH * sizeof(float));
    unsigned int* mx = (unsigned int*)carve((size_t)NN * HH * sizeof(unsigned int));
    float* den      = (float*)carve((size_t)NN * HH * sizeof(float));
    float* hidden   = qbuf;  // alias: MLP hidden is dead before Q projections

    auto gblocks = [](int M) { return dim3((unsigned)((M + 127) / 128)); };
    const int eblocks = (EE * HH + 255) / 256;

    // ---- text MLPs: h = relu(text @ W1 + b1) @ W2 + b2 ----
    gemm_f32_wmma<<<gblocks(N_U), 256, 0, stream>>>(user_text, Wu1, bu1, nullptr, hidden, N_U, TT, 1);
    gemm_f32_wmma<<<gblocks(N_U), 256, 0, stream>>>(hidden, Wu2, bu2, nullptr, text, N_U, DD, 0);
    gemm_f32_wmma<<<gblocks(N_I), 256, 0, stream>>>(item_text, Wi1, bi1, nullptr, hidden, N_I, TT, 1);
    gemm_f32_wmma<<<gblocks(N_I), 256, 0, stream>>>(hidden, Wi2, bi2, nullptr, text + (size_t)N_U * DD, N_I, DD, 0);
    hipMemcpyAsync(out + OFF_HU, text, (size_t)N_U * DD * sizeof(float), hipMemcpyDeviceToDevice, stream);
    hipMemcpyAsync(out + OFF_HI, text + (size_t)N_U * DD, (size_t)N_I * DD * sizeof(float), hipMemcpyDeviceToDevice, stream);

    // ---- ego = concat(user_emb, item_emb) ----
    hipMemcpyAsync(ego, user_emb, (size_t)N_U * DD * sizeof(float), hipMemcpyDeviceToDevice, stream);
    hipMemcpyAsync(ego + (size_t)N_U * DD, item_emb, (size_t)N_I * DD * sizeof(float), hipMemcpyDeviceToDevice, stream);

    auto run_pass = [&](const int* rows, const int* cols, int r, float* dstpart) {
        hipMemsetAsync(mx,  0, (size_t)NN * HH * sizeof(unsigned int), stream);
        hipMemsetAsync(den, 0, (size_t)NN * HH * sizeof(float), stream);
        hipMemsetAsync(agg, 0, NDb, stream);
        edge_scores_kernel <<<eblocks, 256, 0, stream>>>(rows, cols, qbuf, kbuf, rel_emb + r * DD, scores, mx);
        edge_exp_kernel    <<<eblocks, 256, 0, stream>>>(cols, scores, mx, den);
        edge_scatter_kernel<<<eblocks, 256, 0, stream>>>(rows, cols, scores, den, vbuf, agg);
        // out @ Wo[r] + bo[r]
        gemm_f32_wmma<<<gblocks(NN), 256, 0, stream>>>(agg, Wo + (size_t)r * DD * DD, bo + r * DD, nullptr, attnproj, NN, DD, 0);
        // lrelu( (attn_out * rel0[r]) @ W_gc )  ==  lrelu( attn_out @ (diag(rel0[r]) W_gc) )
        gemm_f32_wmma<<<gblocks(NN), 256, 0, stream>>>(attnproj, W_gc, nullptr, rel_emb + r * DD, dstpart, NN, DD, 2);
    };

    // ---- per-relation projections + edge passes (r0/r1 shared across branches) ----
    for (int r = 0; r < RR; ++r) {
        gemm_f32_wmma<<<gblocks(NN), 256, 0, stream>>>(text, Wq + (size_t)r * DD * DD, bq + r * DD, nullptr, qbuf, NN, DD, 0);
        gemm_f32_wmma<<<gblocks(NN), 256, 0, stream>>>(text, Wk + (size_t)r * DD * DD, bk + r * DD, nullptr, kbuf, NN, DD, 0);
        gemm_f32_wmma<<<gblocks(NN), 256, 0, stream>>>(ego,  Wv + (size_t)r * DD * DD, bv + r * DD, nullptr, vbuf, NN, DD, 0);
        run_pass(edges + (size_t)(r * 2 + 0) * EE, edges + (size_t)(r * 2 + 1) * EE, r, part[r]);
        if (r == 2)
            run_pass(aug, aug + EE, 2, part[3]);
    }

    // ---- combine + tail ----
    combine_kernel<<<(int)(((long)NN * DD + 255) / 256), 256, 0, stream>>>(
        ego, part[0], part[1], part[2], part[3], coeff, out);
    tail_kernel<<<1, 192, 0, stream>>>(rel_emb, W_rel, out);
}